// HausdorffLoss_69947837382980
// MI455X (gfx1250) — compile-verified
//
#include <hip/hip_runtime.h>
#include <hip/hip_bf16.h>
#include <stdint.h>

#define BIGF 1e9f
constexpr int B = 4, C = 3, H = 256, W = 256;
constexpr int HW = H * W;
constexpr int NPIX = B * HW;          // 262144
constexpr int CHW = C * HW;           // per-batch logits stride

// ---------------------------------------------------------------------------
// ws layout (float elements):
//   probs : [B,C,H,W]            = 786432 floats
//   predc : [B,H,W]  (int)       = 262144 ints
//   g2    : [4][B][C][H][W]      = 4*786432 floats  (src*2+pol major)
//   anyf  : [2][B][C] (int, pad) = 32 ints
//   part  : [B*H]                = 1024 floats
// ---------------------------------------------------------------------------

__global__ void k_init(int* __restrict__ anyf) {
  if (threadIdx.x < 32) anyf[threadIdx.x] = 0;
}

__global__ void k_classify(const float* __restrict__ logits,
                           const int* __restrict__ targets,
                           float* __restrict__ probs,
                           int* __restrict__ predc,
                           int* __restrict__ anyf) {
  int p = blockIdx.x * blockDim.x + threadIdx.x;
  if (p >= NPIX) return;
  int b  = p / HW;
  int hw = p - b * HW;
  float l0 = logits[b * CHW + 0 * HW + hw];
  float l1 = logits[b * CHW + 1 * HW + hw];
  float l2 = logits[b * CHW + 2 * HW + hw];
  float m  = fmaxf(l0, fmaxf(l1, l2));
  float e0 = expf(l0 - m), e1 = expf(l1 - m), e2 = expf(l2 - m);
  float inv = 1.0f / (e0 + e1 + e2);
  probs[b * CHW + 0 * HW + hw] = e0 * inv;
  probs[b * CHW + 1 * HW + hw] = e1 * inv;
  probs[b * CHW + 2 * HW + hw] = e2 * inv;
  // argmax over classes, first-max on ties (matches jnp.argmax)
  int a = 0; float bm = l0;
  if (l1 > bm) { bm = l1; a = 1; }
  if (l2 > bm) { bm = l2; a = 2; }
  predc[p] = a;
  int t = targets[p];
  // benign races: everyone stores 1
  anyf[0 * B * C + b * C + t] = 1;
  anyf[1 * B * C + b * C + a] = 1;
}

// One block per EDT image (48 = 2 src * 2 pol * B * C), one thread per column.
// Two sequential scans over H replicating the reference's
// d = is_zero ? 0 : min(carry + 1, BIG); then g = min(fwd,bwd); store g^2.
__global__ void k_edt_vertical(const int* __restrict__ targets,
                               const int* __restrict__ predc,
                               float* __restrict__ g2) {
  int img = blockIdx.x;            // ((src*2+pol)*B + b)*C + c
  int w   = threadIdx.x;
  int src = img / 24;
  int rem = img % 24;
  int pol = rem / 12;
  int bc  = rem % 12;
  int b = bc / C, c = bc % C;
  const int* lab = (src == 0) ? targets : predc;
  const int* col = lab + b * HW + w;
  float* out = g2 + (size_t)img * HW + w;

  float carry = BIGF;
#pragma unroll 4
  for (int h = 0; h < H; ++h) {
    int cls  = col[h * W];
    bool isz = (pol == 0) ? (cls != c) : (cls == c);  // zero pixel of mask
    float d  = isz ? 0.0f : fminf(carry + 1.0f, BIGF);
    carry = d;
    out[h * W] = d;
  }
  carry = BIGF;
#pragma unroll 4
  for (int h = H - 1; h >= 0; --h) {
    int cls  = col[h * W];
    bool isz = (pol == 0) ? (cls != c) : (cls == c);
    float d  = isz ? 0.0f : fminf(carry + 1.0f, BIGF);
    carry = d;
    float g = fminf(d, out[h * W]);   // same-thread store->load, in order
    out[h * W] = g * g;
  }
}

// One block per (b,h) row. Stage 12 g^2 rows into LDS *transposed*
// (rowsT[jp][q], 48 B per jp) with CDNA5 async global->LDS scatter copies,
// then the exact min-plus over columns reads 3x ds_load_b128 per step
// (broadcast, conflict-free) instead of 12x b32.
__global__ void k_edt_horiz_accum(const float* __restrict__ g2,
                                  const float* __restrict__ probs,
                                  const int* __restrict__ targets,
                                  const int* __restrict__ anyf,
                                  float* __restrict__ part) {
  __shared__ __align__(16) float rowsT[W * 12];   // 12 KB, [jp][q]
  __shared__ float red[256];
  int r = blockIdx.x;              // [0, B*H)
  int b = r / H, h = r % H;
  int j = threadIdx.x;

#if defined(__HIP_DEVICE_COMPILE__)
  // Lane j async-copies element jp=j of each of the 12 rows into rowsT[j][q].
  for (int q = 0; q < 12; ++q) {   // q = (src*2+pol)*3 + c
    int sp = q / 3, c = q % 3;
    const float* gp = g2 + (size_t)((sp * B + b) * C + c) * HW + h * W + j;
    unsigned ldsoff = (unsigned)(size_t)&rowsT[j * 12 + q];
    unsigned long long ga = (unsigned long long)(size_t)gp;
    asm volatile("global_load_async_to_lds_b32 %0, %1, off"
                 :: "v"(ldsoff), "v"(ga) : "memory");
  }
  asm volatile("s_wait_asynccnt 0x0" ::: "memory");
#endif
  __syncthreads();

  float d2[12];
#pragma unroll
  for (int q = 0; q < 12; ++q) d2[q] = 3.4e38f;
  float jf = (float)j;
  const float4* rp = (const float4*)(&rowsT[0]);   // 3 float4 per jp
#pragma unroll 4
  for (int jp = 0; jp < W; ++jp) {
    float off  = jf - (float)jp;
    float off2 = off * off;
    float4 v0 = rp[jp * 3 + 0];
    float4 v1 = rp[jp * 3 + 1];
    float4 v2 = rp[jp * 3 + 2];
    d2[0]  = fminf(d2[0],  v0.x + off2);
    d2[1]  = fminf(d2[1],  v0.y + off2);
    d2[2]  = fminf(d2[2],  v0.z + off2);
    d2[3]  = fminf(d2[3],  v0.w + off2);
    d2[4]  = fminf(d2[4],  v1.x + off2);
    d2[5]  = fminf(d2[5],  v1.y + off2);
    d2[6]  = fminf(d2[6],  v1.z + off2);
    d2[7]  = fminf(d2[7],  v1.w + off2);
    d2[8]  = fminf(d2[8],  v2.x + off2);
    d2[9]  = fminf(d2[9],  v2.y + off2);
    d2[10] = fminf(d2[10], v2.z + off2);
    d2[11] = fminf(d2[11], v2.w + off2);
  }

  int tcls = targets[b * HW + h * W + j];
  float acc = 0.0f;
#pragma unroll
  for (int c = 0; c < C; ++c) {
    float tdm = (anyf[b * C + c] != 0)
                    ? (sqrtf(d2[0 * 3 + c]) + sqrtf(d2[1 * 3 + c])) : 0.0f;
    float pdm = (anyf[B * C + b * C + c] != 0)
                    ? (sqrtf(d2[2 * 3 + c]) + sqrtf(d2[3 * 3 + c])) : 0.0f;
    float dist = tdm * tdm + pdm * pdm;       // alpha = 2
    float pr   = probs[b * CHW + c * HW + h * W + j];
    float err  = pr - ((tcls == c) ? 1.0f : 0.0f);
    acc += err * err * dist;
  }
  red[j] = acc;
  __syncthreads();
  for (int o = 128; o > 0; o >>= 1) {
    if (j < o) red[j] += red[j + o];
    __syncthreads();
  }
  if (j == 0) part[r] = red[0];
}

__global__ void k_final(const float* __restrict__ part, float* __restrict__ out) {
  __shared__ float red[256];
  int t = threadIdx.x;
  float s = part[t] + part[t + 256] + part[t + 512] + part[t + 768];
  red[t] = s;
  __syncthreads();
  for (int o = 128; o > 0; o >>= 1) {
    if (t < o) red[t] += red[t + o];
    __syncthreads();
  }
  if (t == 0) out[0] = red[0] / (float)(B * C * HW);
}

extern "C" void kernel_launch(void* const* d_in, const int* in_sizes, int n_in,
                              void* d_out, int out_size, void* d_ws, size_t ws_size,
                              hipStream_t stream) {
  const float* logits  = (const float*)d_in[0];
  const int*   targets = (const int*)d_in[1];

  float* ws    = (float*)d_ws;
  float* probs = ws;                                   // 786432 f
  int*   predc = (int*)(ws + 786432);                  // 262144 i
  float* g2    = ws + 786432 + 262144;                 // 3145728 f
  int*   anyf  = (int*)(g2 + 4 * 786432);              // 32 i
  float* part  = (float*)(anyf + 32);                  // 1024 f
  float* out   = (float*)d_out;

  hipLaunchKernelGGL(k_init, dim3(1), dim3(32), 0, stream, anyf);
  hipLaunchKernelGGL(k_classify, dim3(NPIX / 256), dim3(256), 0, stream,
                     logits, targets, probs, predc, anyf);
  hipLaunchKernelGGL(k_edt_vertical, dim3(48), dim3(256), 0, stream,
                     targets, predc, g2);
  hipLaunchKernelGGL(k_edt_horiz_accum, dim3(B * H), dim3(256), 0, stream,
                     g2, probs, targets, anyf, part);
  hipLaunchKernelGGL(k_final, dim3(1), dim3(256), 0, stream, part, out);
}